// MPS_23897198035350
// MI455X (gfx1250) — compile-verified
//
#include <hip/hip_runtime.h>

#define B_    8192
#define N_    128
#define D_    128
#define R_    64
#define NSTEP 126   // N_ - 2

typedef __attribute__((ext_vector_type(2))) float v2f;
typedef __attribute__((ext_vector_type(8))) float v8f;

#if defined(__has_builtin)
#if __has_builtin(__builtin_amdgcn_global_load_async_to_lds_b128)
#define HAS_ASYNC_LDS 1
#endif
#if __has_builtin(__builtin_amdgcn_s_wait_asynccnt)
#define HAS_WAIT_ASYNC_BUILTIN 1
#endif
#endif

#ifdef HAS_ASYNC_LDS
// Builtin params are pointers to 16-byte int vectors in AS(1)/AS(3)
// (diagnostic: "__attribute__((__vector_size__(4 * sizeof(int)))) int __device__ *").
typedef int v4i_vs __attribute__((vector_size(16)));
typedef __attribute__((address_space(1))) v4i_vs* gv4p_t;
typedef __attribute__((address_space(3))) v4i_vs* lv4p_t;
#endif

__device__ __forceinline__ void wait_asynccnt0() {
#ifdef HAS_ASYNC_LDS
#ifdef HAS_WAIT_ASYNC_BUILTIN
    __builtin_amdgcn_s_wait_asynccnt(0);
#else
    asm volatile("s_wait_asynccnt 0x0" ::: "memory");
#endif
#endif
}

// ---------------------------------------------------------------------------
// Kernel 1: per-step counting sort of batch rows by x[:, s+1].
// One block per step. Output: perm[s*B_ + pos] = b, boff[s*(D_+1) + d] offsets.
// ---------------------------------------------------------------------------
__global__ void mps_perm_kernel(const int* __restrict__ x,
                                int* __restrict__ perm,
                                int* __restrict__ boff) {
    __shared__ int hist[D_];
    __shared__ int base[D_ + 1];
    const int s   = blockIdx.x;      // step 0..NSTEP-1 -> uses x[:, s+1]
    const int tid = threadIdx.x;

    for (int d = tid; d < D_; d += blockDim.x) hist[d] = 0;
    __syncthreads();
    for (int b = tid; b < B_; b += blockDim.x)
        atomicAdd(&hist[x[b * N_ + (s + 1)]], 1);
    __syncthreads();
    if (tid == 0) {
        int acc = 0;
        for (int d = 0; d < D_; ++d) { base[d] = acc; acc += hist[d]; }
        base[D_] = acc;
    }
    __syncthreads();
    for (int d = tid; d <= D_; d += blockDim.x)
        boff[s * (D_ + 1) + d] = base[d];
    for (int d = tid; d < D_; d += blockDim.x) hist[d] = base[d];
    __syncthreads();
    for (int b = tid; b < B_; b += blockDim.x) {
        int d   = x[b * N_ + (s + 1)];
        int pos = atomicAdd(&hist[d], 1);
        perm[s * B_ + pos] = b;
    }
}

// ---------------------------------------------------------------------------
// Kernel 2: psi0[b, r] = g_first[0, x[b,0], r]
// ---------------------------------------------------------------------------
__global__ void mps_init_kernel(const int* __restrict__ x,
                                const float* __restrict__ g_first,
                                float* __restrict__ psi) {
    int t = blockIdx.x * blockDim.x + threadIdx.x;   // B_*R_ threads
    int b = t >> 6;
    int r = t & (R_ - 1);
    psi[t] = g_first[x[b * N_] * R_ + r];
}

// ---------------------------------------------------------------------------
// Kernel 3: one step. Block = one bucket d. 128 threads = 4 waves.
// psi_out[rows] = psi_in[rows] x G_d, G_d[k][q] = g_mid_step[k*D_*R_ + d*R_ + q]
// WMMA f32 16x16x4, K=64 chained (exact f32 math; problem is memory-bound).
// G_d staged via async global->LDS (ASYNCcnt) when available.
// ---------------------------------------------------------------------------
#define GS_STRIDE 68   // 64 + 4: keeps 16B alignment, kills A-fragment conflicts

__global__ void __launch_bounds__(128)
mps_step_kernel(const float* __restrict__ psi_in,
                float* __restrict__ psi_out,
                const float* __restrict__ g_mid_step,   // + step*R_*D_*R_
                const int* __restrict__ perm_s,         // + s*B_
                const int* __restrict__ boff_s) {       // + s*(D_+1)
    __shared__ float Gs[R_ * GS_STRIDE];      // [k][q]
    __shared__ float Ps[16 * GS_STRIDE];      // [row][k]

    const int d     = blockIdx.x;
    const int start = boff_s[d];
    const int cnt   = boff_s[d + 1] - start;
    if (cnt == 0) return;                      // block-uniform

    const int tid  = threadIdx.x;
    const int wave = tid >> 5;                 // 0..3 -> output col tile
    const int lane = tid & 31;
    const int hl   = lane >> 4;                // half-wave 0/1
    const int l16  = lane & 15;

    // Stage G_d into LDS: 64 rows x 64 f32 (16 KB), 128 threads x 8 chunks.
    const float* gsrc = g_mid_step + (size_t)d * R_;
#ifdef HAS_ASYNC_LDS
    for (int e = tid; e < R_ * (R_ / 4); e += 128) {
        int k  = e >> 4;                       // row (contraction index)
        int q4 = e & 15;                       // float4 within row
        __builtin_amdgcn_global_load_async_to_lds_b128(
            (gv4p_t)(gsrc + (size_t)k * (D_ * R_) + q4 * 4),
            (lv4p_t)&Gs[k * GS_STRIDE + q4 * 4],
            /*offset=*/0, /*cpol=*/0);
    }
    wait_asynccnt0();                          // own async copies landed in LDS
#else
    for (int e = tid; e < R_ * (R_ / 4); e += 128) {
        int k  = e >> 4;
        int q4 = e & 15;
        *(float4*)&Gs[k * GS_STRIDE + q4 * 4] =
            *(const float4*)(gsrc + (size_t)k * (D_ * R_) + q4 * 4);
    }
#endif

    const int ntiles = (cnt + 15) >> 4;
    for (int t = 0; t < ntiles; ++t) {
        // Stage 16 gathered psi rows: 8 threads/row, 8 floats each.
        {
            int row  = tid >> 3;               // 0..15
            int part = tid & 7;                // 0..7
            int gr   = t * 16 + row;
            float4 a0 = make_float4(0.f, 0.f, 0.f, 0.f);
            float4 a1 = a0;
            if (gr < cnt) {
                int bidx = perm_s[start + gr];
                const float* p = psi_in + (size_t)bidx * R_ + part * 8;
                a0 = *(const float4*)p;
                a1 = *(const float4*)(p + 4);
            }
            *(float4*)&Ps[row * GS_STRIDE + part * 8]     = a0;
            *(float4*)&Ps[row * GS_STRIDE + part * 8 + 4] = a1;
        }
        __syncthreads();                       // publishes Gs (1st iter) + Ps

        // 16x64 @ 64x16 via 16 chained v_wmma_f32_16x16x4_f32.
        v8f acc = {};
#pragma unroll
        for (int k0 = 0; k0 < R_; k0 += 4) {
            int ka = k0 + hl * 2;
            v2f a, b;
            // A 16x4: lane<16 -> K={k0,k0+1}, lane>=16 -> K={k0+2,k0+3}, M=l16
            a.x = Ps[l16 * GS_STRIDE + ka];
            a.y = Ps[l16 * GS_STRIDE + ka + 1];
            // B 4x16: same K split, N = wave*16 + l16
            b.x = Gs[ka * GS_STRIDE + wave * 16 + l16];
            b.y = Gs[(ka + 1) * GS_STRIDE + wave * 16 + l16];
            acc = __builtin_amdgcn_wmma_f32_16x16x4_f32(
                /*neg_a=*/false, a, /*neg_b=*/false, b,
                /*c_mod=*/(short)0, acc, /*reuse_a=*/false, /*reuse_b=*/false);
        }

        // D 16x16: VGPR j -> row j (lanes 0-15) / j+8 (lanes 16-31), col l16.
#pragma unroll
        for (int j = 0; j < 8; ++j) {
            int row = j + hl * 8;
            int gr  = t * 16 + row;
            if (gr < cnt) {
                int bidx = perm_s[start + gr];
                psi_out[(size_t)bidx * R_ + wave * 16 + l16] = acc[j];
            }
        }
        __syncthreads();
    }
}

// ---------------------------------------------------------------------------
// Kernel 4: out[b] = sum_r psi[b,r] * g_last[r, x[b,N-1], 0]
// ---------------------------------------------------------------------------
__global__ void mps_final_kernel(const float* __restrict__ psi,
                                 const int* __restrict__ x,
                                 const float* __restrict__ g_last,
                                 float* __restrict__ out) {
    int b = blockIdx.x * blockDim.x + threadIdx.x;
    if (b >= B_) return;
    int xb = x[b * N_ + (N_ - 1)];
    const float* p = psi + (size_t)b * R_;
    float s = 0.f;
#pragma unroll 8
    for (int r = 0; r < R_; ++r)
        s += p[r] * g_last[r * D_ + xb];
    out[b] = s;
}

// ---------------------------------------------------------------------------
extern "C" void kernel_launch(void* const* d_in, const int* in_sizes, int n_in,
                              void* d_out, int out_size, void* d_ws, size_t ws_size,
                              hipStream_t stream) {
    const int*   x      = (const int*)d_in[0];
    const float* gfirst = (const float*)d_in[1];
    const float* gmid   = (const float*)d_in[2];
    const float* glast  = (const float*)d_in[3];
    float* out = (float*)d_out;

    char* ws = (char*)d_ws;
    float* psiA = (float*)ws;                                          // B_*R_ f32
    float* psiB = (float*)(ws + (size_t)B_ * R_ * sizeof(float));      // B_*R_ f32
    int*   perm = (int*)(ws + (size_t)2 * B_ * R_ * sizeof(float));    // NSTEP*B_
    int*   boff = (int*)(ws + (size_t)2 * B_ * R_ * sizeof(float)
                            + (size_t)NSTEP * B_ * sizeof(int));       // NSTEP*(D_+1)

    mps_perm_kernel<<<NSTEP, 256, 0, stream>>>(x, perm, boff);
    mps_init_kernel<<<(B_ * R_) / 256, 256, 0, stream>>>(x, gfirst, psiA);

    for (int s = 0; s < NSTEP; ++s) {
        const float* pin  = (s & 1) ? psiB : psiA;
        float*       pout = (s & 1) ? psiA : psiB;
        mps_step_kernel<<<D_, 128, 0, stream>>>(
            pin, pout,
            gmid + (size_t)s * R_ * D_ * R_,
            perm + (size_t)s * B_,
            boff + (size_t)s * (D_ + 1));
    }
    // NSTEP = 126 (even) -> final psi lives in psiA
    mps_final_kernel<<<B_ / 256, 256, 0, stream>>>(psiA, x, glast, out);
}